// NMS_79534204388019
// MI455X (gfx1250) — compile-verified
//
#include <hip/hip_runtime.h>
#include <hip/hip_bf16.h>
#include <stdint.h>

// Problem constants (match reference)
#define B_        32
#define N_        8400
#define C_        80
#define MAXDET    100
#define SCORE_THR 0.6f
#define IOU_THR   0.5f

#define TPB 1024           // 32 waves (wave32)
#define KPT 9              // ceil(8400/1024) boxes+scores per thread in registers

typedef __attribute__((ext_vector_type(4))) unsigned int u32x4;
typedef __attribute__((ext_vector_type(8))) unsigned int u32x8;

// ---------------------------------------------------------------------------
// TDM: DMA a strided column (tile_dim0 = 1 element, tile_dim1 = rows) from a
// row-major [rows x row_len] f32 tensor into LDS. CDNA5 tensor_load_to_lds,
// 2D form (D# groups 2/3 NULL). Tracked by TENSORcnt.
// ---------------------------------------------------------------------------
__device__ __forceinline__ void tdm_load_strided_col(unsigned lds_byte_off,
                                                     const float* gptr,
                                                     unsigned rows,
                                                     unsigned row_len_elems) {
  unsigned long long ga = (unsigned long long)(uintptr_t)gptr;
  u32x4 g0;
  g0.x = 1u;                                          // count=1, user mode
  g0.y = lds_byte_off;                                // lds_addr [63:32]
  g0.z = (unsigned)ga;                                // global_addr[31:0]
  g0.w = ((unsigned)(ga >> 32) & 0x01FFFFFFu)         // global_addr[56:32]
       | (2u << 30);                                  // type = 2 ("image")
  u32x8 g1;
  g1.s0 = (2u << 16);                                 // data_size = 4B
  g1.s1 = (row_len_elems & 0xFFFFu) << 16;            // tensor_dim0[15:0]
  g1.s2 = ((row_len_elems >> 16) & 0xFFFFu)           // tensor_dim0[31:16]
        | ((rows & 0xFFFFu) << 16);                   // tensor_dim1[15:0]
  g1.s3 = ((rows >> 16) & 0xFFFFu)                    // tensor_dim1[31:16]
        | (1u << 16);                                 // tile_dim0 = 1 elem
  g1.s4 = rows & 0xFFFFu;                             // tile_dim1 = rows, tile_dim2=0
  g1.s5 = row_len_elems;                              // tensor_dim0_stride[31:0]
  g1.s6 = 0u;
  g1.s7 = 0u;
  asm volatile("tensor_load_to_lds %0, %1" :: "s"(g0), "s"(g1) : "memory");
}

// ---------------------------------------------------------------------------
// Phase 1: per-(class, batch) greedy NMS. blockIdx.x = class, blockIdx.y = b.
// Emits up to MAXDET survivors (descending score) + count into workspace.
// ---------------------------------------------------------------------------
__global__ __launch_bounds__(TPB) void nms_per_class(
    const float* __restrict__ boxes,   // [B, N, 4]
    const float* __restrict__ scores,  // [B, N, C]
    float* __restrict__ ws_score,      // [B, C, MAXDET]
    int*   __restrict__ ws_idx,        // [B, C, MAXDET]
    int*   __restrict__ ws_cnt) {      // [B, C]
  const int c   = blockIdx.x;
  const int b   = blockIdx.y;
  const int tid = threadIdx.x;
  const int wid = tid >> 5;
  const int lane = tid & 31;

  const float4* bx4 = (const float4*)(boxes + (size_t)b * N_ * 4);

  __shared__ float lds_sc[N_];      // class score column, filled by TDM
  __shared__ float red_s[32];
  __shared__ int   red_i[32];
  __shared__ float bcast[6];        // [best_score, best_idx(bits), x1,y1,x2,y2]

  if (tid == 0) {
    const float* col = scores + (size_t)b * N_ * C_ + c;  // stride C_ elements
    tdm_load_strided_col((unsigned)(uintptr_t)(&lds_sc[0]), col, N_, C_);
  }
  __builtin_amdgcn_s_wait_tensorcnt(0);
  __syncthreads();

  // Pull this thread's 9 scores (thresholded) + 9 boxes into registers.
  float  s[KPT];
  float4 mb[KPT];
#pragma unroll
  for (int k = 0; k < KPT; ++k) {
    const int n = tid + k * TPB;
    if (n < N_) {
      const float v = lds_sc[n];
      s[k]  = (v > SCORE_THR) ? v : -1.0f;
      mb[k] = bx4[n];
    } else {
      s[k]  = -1.0f;
      mb[k] = make_float4(0.f, 0.f, 0.f, 0.f);
    }
  }

  int nacc = 0;
  for (int it = 0; it < MAXDET; ++it) {
    // --- local argmax over registers ---
    float bs = -1.0f;
    int   bn = tid;
#pragma unroll
    for (int k = 0; k < KPT; ++k) {
      const int n = tid + k * TPB;
      if (s[k] > bs) { bs = s[k]; bn = n; }
    }
    // --- wave32 butterfly reduce (prefer smaller index on tie) ---
#pragma unroll
    for (int off = 16; off > 0; off >>= 1) {
      const float os = __shfl_xor(bs, off, 32);
      const int   on = __shfl_xor(bn, off, 32);
      if (os > bs || (os == bs && on < bn)) { bs = os; bn = on; }
    }
    if (lane == 0) { red_s[wid] = bs; red_i[wid] = bn; }
    __syncthreads();
    // --- cross-wave reduce by wave 0 ---
    if (wid == 0) {
      bs = red_s[lane];
      bn = red_i[lane];
#pragma unroll
      for (int off = 16; off > 0; off >>= 1) {
        const float os = __shfl_xor(bs, off, 32);
        const int   on = __shfl_xor(bn, off, 32);
        if (os > bs || (os == bs && on < bn)) { bs = os; bn = on; }
      }
      if (lane == 0) {
        bcast[0] = bs;
        bcast[1] = __int_as_float(bn);
        if (bs > SCORE_THR) {
          const float4 bb = bx4[bn];
          bcast[2] = bb.x; bcast[3] = bb.y; bcast[4] = bb.z; bcast[5] = bb.w;
        }
      }
    }
    __syncthreads();
    bs = bcast[0];
    bn = __float_as_int(bcast[1]);
    if (bs <= SCORE_THR) break;   // uniform across block

    if (tid == 0) {
      const int base = (b * C_ + c) * MAXDET + nacc;
      ws_score[base] = bs;
      ws_idx[base]   = bn;
    }
    ++nacc;

    // --- suppress same-class boxes with IoU > thr (division-free) ---
    const float bx1 = bcast[2], by1 = bcast[3], bx2 = bcast[4], by2 = bcast[5];
    const float bArea = fmaxf(bx2 - bx1, 0.f) * fmaxf(by2 - by1, 0.f);
#pragma unroll
    for (int k = 0; k < KPT; ++k) {
      const float lx = fmaxf(bx1, mb[k].x), ly = fmaxf(by1, mb[k].y);
      const float rx = fminf(bx2, mb[k].z), ry = fminf(by2, mb[k].w);
      const float inter = fmaxf(rx - lx, 0.f) * fmaxf(ry - ly, 0.f);
      const float area  = fmaxf(mb[k].z - mb[k].x, 0.f) * fmaxf(mb[k].w - mb[k].y, 0.f);
      const float uni   = bArea + area - inter;
      if (inter > IOU_THR * (uni + 1e-9f)) s[k] = -1.0f;  // iou > IOU_THR
    }
  }
  if (tid == 0) ws_cnt[b * C_ + c] = nacc;
}

// ---------------------------------------------------------------------------
// Phase 2: per-batch 80-way merge of descending per-class lists -> top MAXDET.
// One wave32 per batch. Output layout (flat, reference return order):
//   boxes  [B,100,4] f32 | scores [B,100] f32 | labels [B,100] i32 | n_valid [B,1] i32
// ---------------------------------------------------------------------------
__global__ __launch_bounds__(32) void nms_merge(
    const float* __restrict__ boxes,
    const float* __restrict__ ws_score,
    const int*   __restrict__ ws_idx,
    const int*   __restrict__ ws_cnt,
    float* __restrict__ outF,
    int*   __restrict__ outI) {
  const int b    = blockIdx.x;
  const int lane = threadIdx.x;

  __shared__ int head[C_];
  __shared__ int cnts[C_];
  for (int ci = lane; ci < C_; ci += 32) { head[ci] = 0; cnts[ci] = ws_cnt[b * C_ + ci]; }

  float* obox = outF + (size_t)b * MAXDET * 4;
  float* oscr = outF + (size_t)B_ * MAXDET * 4 + (size_t)b * MAXDET;
  int*   olab = outI + (size_t)B_ * MAXDET * 5 + (size_t)b * MAXDET;
  for (int i = lane; i < MAXDET * 4; i += 32) obox[i] = 0.f;
  for (int i = lane; i < MAXDET; i += 32) { oscr[i] = 0.f; olab[i] = 0; }
  __syncthreads();

  const float4* bx4 = (const float4*)(boxes + (size_t)b * N_ * 4);

  int ndet = 0;
  for (int it = 0; it < MAXDET; ++it) {
    float bs = -1.0f;
    int   bc = -1;
    for (int ci = lane; ci < C_; ci += 32) {
      const int h = head[ci];
      if (h < cnts[ci]) {
        const float v = ws_score[(size_t)(b * C_ + ci) * MAXDET + h];
        if (v > bs) { bs = v; bc = ci; }
      }
    }
#pragma unroll
    for (int off = 16; off > 0; off >>= 1) {
      const float os = __shfl_xor(bs, off, 32);
      const int   oc = __shfl_xor(bc, off, 32);
      if (os > bs || (os == bs && oc != -1 && (bc == -1 || oc < bc))) { bs = os; bc = oc; }
    }
    if (bc < 0) break;   // uniform: all class lists exhausted

    if (lane == 0) {
      const int h = head[bc];
      const int n = ws_idx[(size_t)(b * C_ + bc) * MAXDET + h];
      const float4 bb = bx4[n];
      obox[it * 4 + 0] = bb.x;
      obox[it * 4 + 1] = bb.y;
      obox[it * 4 + 2] = bb.z;
      obox[it * 4 + 3] = bb.w;
      oscr[it] = bs;
      olab[it] = bc;
      head[bc] = h + 1;
    }
    ++ndet;
    __syncthreads();  // single-wave WG: LDS in-order, barrier is cheap/NOP
  }
  if (lane == 0) outI[(size_t)B_ * MAXDET * 6 + b] = ndet;  // n_valid
}

// ---------------------------------------------------------------------------
extern "C" void kernel_launch(void* const* d_in, const int* in_sizes, int n_in,
                              void* d_out, int out_size, void* d_ws, size_t ws_size,
                              hipStream_t stream) {
  (void)in_sizes; (void)n_in; (void)out_size; (void)ws_size;
  const float* boxes  = (const float*)d_in[0];   // [B,N,4]
  const float* scores = (const float*)d_in[1];   // [B,N,C]

  // Workspace: survivors per (batch, class) + counts (~2.06 MB)
  float* ws_score = (float*)d_ws;
  int*   ws_idx   = (int*)((char*)d_ws + sizeof(float) * (size_t)B_ * C_ * MAXDET);
  int*   ws_cnt   = (int*)((char*)d_ws + (sizeof(float) + sizeof(int)) * (size_t)B_ * C_ * MAXDET);

  nms_per_class<<<dim3(C_, B_), TPB, 0, stream>>>(boxes, scores, ws_score, ws_idx, ws_cnt);
  nms_merge<<<dim3(B_), 32, 0, stream>>>(boxes, ws_score, ws_idx, ws_cnt,
                                         (float*)d_out, (int*)d_out);
}